// rits_16647293239724
// MI455X (gfx1250) — compile-verified
//
#include <hip/hip_runtime.h>

// Problem constants (from reference)
#define BDIM 128
#define TDIM 256
#define FDIM 36
#define HDIM 128
#define K3   3
#define BB   16            // batch rows per workgroup
#define RROWS 48           // BB * K3 LSTM rows per workgroup
#define NBLK 8             // BDIM / BB
#define H4   512           // 4*H gate width
#define KIN  200           // 2F + H = 72 + 128

// Workspace layout (float offsets)
#define WS_WT     0                          // [KIN*H4]   = 102400  (concat [W_ih|W_hh] transposed)
#define WS_WDHT   (WS_WT + KIN*H4)           // [F*H]      = 4608    W_dh^T  [f][h]
#define WS_WHISTT (WS_WDHT + FDIM*HDIM)      // [H*F]      = 4608    W_hist^T [h][f]
#define WS_WFT    (WS_WHISTT + HDIM*FDIM)    // [F*F]      = 1296    (W_feat*(1-I))^T [f][g]
#define WS_WWCT   (WS_WFT + FDIM*FDIM)       // [2F*F]     = 2592    W_wc^T [f2][g]
#define WS_LOSS   (WS_WWCT + 2*FDIM*FDIM)    // [T*NBLK*4] = 8192    per-(t,block) loss partials

typedef __attribute__((ext_vector_type(2))) float v2f;
typedef __attribute__((ext_vector_type(8))) float v8f;

static __device__ __forceinline__ v8f wmma4(v2f a, v2f b, v8f c) {
  // D = A(16x4 f32) * B(4x16 f32) + C(16x16 f32), wave32, exact fp32
  return __builtin_amdgcn_wmma_f32_16x16x4_f32(false, a, false, b, (short)0, c, false, false);
}
static __device__ __forceinline__ float sigf(float x) { return 1.0f / (1.0f + __expf(-x)); }

// ---------------------------------------------------------------------------
// Kernel 0: transpose weights into workspace (runs once per launch, L2-hot)
// ---------------------------------------------------------------------------
__global__ void rits_prep_kernel(const float* __restrict__ W_dh,
                                 const float* __restrict__ W_hist,
                                 const float* __restrict__ W_feat,
                                 const float* __restrict__ W_wc,
                                 const float* __restrict__ W_ih,
                                 const float* __restrict__ W_hh,
                                 float* __restrict__ ws) {
  int tid = blockIdx.x * blockDim.x + threadIdx.x;
  if (tid < KIN * H4) {                      // combined gates weight, [j][n]
    int j = tid / H4, n = tid % H4;
    ws[WS_WT + tid] = (j < 2 * FDIM) ? W_ih[n * (2 * FDIM) + j]
                                     : W_hh[n * HDIM + (j - 2 * FDIM)];
  }
  if (tid < FDIM * HDIM) {                   // W_dh^T: [f][h]
    int f = tid / HDIM, h2 = tid % HDIM;
    ws[WS_WDHT + tid] = W_dh[h2 * FDIM + f];
  }
  if (tid < HDIM * FDIM) {                   // W_hist^T: [h][f]
    int h2 = tid / FDIM, f = tid % FDIM;
    ws[WS_WHISTT + tid] = W_hist[f * HDIM + h2];
  }
  if (tid < FDIM * FDIM) {                   // masked W_feat^T: [f][g], zero diagonal
    int f = tid / FDIM, g = tid % FDIM;
    ws[WS_WFT + tid] = (f == g) ? 0.0f : W_feat[g * FDIM + f];
  }
  if (tid < 2 * FDIM * FDIM) {               // W_wc^T: [f2][g]
    int f2 = tid / FDIM, g = tid % FDIM;
    ws[WS_WWCT + tid] = W_wc[g * (2 * FDIM) + f2];
  }
}

// ---------------------------------------------------------------------------
// Kernel 1: persistent scan. One workgroup = 16 batch rows, all 256 steps.
// ---------------------------------------------------------------------------
__global__ __launch_bounds__(256) void rits_scan_kernel(
    const float* __restrict__ data,
    const float* __restrict__ b_dh, const float* __restrict__ W_dx,
    const float* __restrict__ b_dx, const float* __restrict__ b_hist,
    const float* __restrict__ b_feat, const float* __restrict__ b_wc,
    const float* __restrict__ b_ih, const float* __restrict__ b_hh,
    const float* __restrict__ W_comb,
    const float* __restrict__ ws, float* __restrict__ lossPart,
    float* __restrict__ out_imp) {
  // LDS-resident state & staging (~146 KB)
  __shared__ float hA[RROWS * HDIM], hB[RROWS * HDIM], c_s[RROWS * HDIM];
  __shared__ float gh[BB * HDIM];
  __shared__ float gx[BB * FDIM], m_s[BB * FDIM], d_s[BB * FDIM];
  __shared__ float rt_s[RROWS * FDIM], tr_s[RROWS * FDIM], se_s[RROWS * FDIM];
  __shared__ float xh_s[RROWS * FDIM], xc_s[RROWS * FDIM], zh_s[RROWS * FDIM];
  __shared__ float al_s[RROWS * FDIM], cc_s[RROWS * FDIM];
  __shared__ float sb_dh[HDIM], sb_hist[FDIM], sb_feat[FDIM], sb_wc[FDIM];
  __shared__ float sb_ih[H4], sb_hh[H4];
  __shared__ float swdx[FDIM], sbdx[FDIM];
  __shared__ float red_s[32];

  const int tid  = threadIdx.x;
  const int wv   = tid >> 5;
  const int lane = tid & 31;
  const int half = lane >> 4;     // which K-pair half of the wave
  const int l16  = lane & 15;
  const int blk  = blockIdx.x;

  // softmax(W_comb) -> wk (sums to 1)
  float c0 = W_comb[0], c1 = W_comb[1], c2 = W_comb[2];
  float mx = fmaxf(c0, fmaxf(c1, c2));
  float e0 = __expf(c0 - mx), e1 = __expf(c1 - mx), e2 = __expf(c2 - mx);
  float es = e0 + e1 + e2;
  const float wk0 = e0 / es, wk1 = e1 / es, wk2 = e2 / es;
  const float wksum = wk0 + wk1 + wk2;

  const float* wsWT     = ws + WS_WT;
  const float* wsWdhT   = ws + WS_WDHT;
  const float* wsWhistT = ws + WS_WHISTT;
  const float* wsWfT    = ws + WS_WFT;
  const float* wsWwcT   = ws + WS_WWCT;

  // init biases / diag / state
  for (int i = tid; i < HDIM; i += 256) sb_dh[i] = b_dh[i];
  for (int i = tid; i < FDIM; i += 256) {
    sb_hist[i] = b_hist[i]; sb_feat[i] = b_feat[i]; sb_wc[i] = b_wc[i];
    swdx[i] = W_dx[i * FDIM + i]; sbdx[i] = b_dx[i];
  }
  for (int i = tid; i < H4; i += 256) { sb_ih[i] = b_ih[i]; sb_hh[i] = b_hh[i]; }
  for (int i = tid; i < RROWS * HDIM; i += 256) { hA[i] = 0.0f; hB[i] = 0.0f; c_s[i] = 0.0f; }
  __syncthreads();

  for (int t = 0; t < TDIM; ++t) {
    float* hR = (t & 1) ? hB : hA;   // read (decayed in place)
    float* hW = (t & 1) ? hA : hB;   // write (new LSTM h)
    float pl1 = 0.0f, pl2 = 0.0f, pl3 = 0.0f, pden = 0.0f;

    // ---- P0: stage this step's inputs; prefetch next step ----
    for (int idx = tid; idx < BB * FDIM; idx += 256) {
      int bb = idx / FDIM, f = idx % FDIM;
      int b = blk * BB + bb;
      size_t base = ((size_t)(b * 24 + 1) * TDIM + t) * FDIM + f;
      m_s[idx] = data[base];
      d_s[idx] = data[base + (size_t)TDIM * FDIM];   // channel 2
      if (t + 1 < TDIM) __builtin_prefetch(&data[base + FDIM], 0, 1);
    }
    for (int idx = tid; idx < RROWS * FDIM; idx += 256) {
      int row = idx / FDIM, f = idx % FDIM;
      int bb = row / K3, k = row % K3;
      int b = blk * BB + bb;
      size_t base = ((size_t)(b * 24 + 3 + 3 * k) * TDIM + t) * FDIM + f;
      rt_s[idx] = data[base];
      tr_s[idx] = data[base + (size_t)TDIM * FDIM];
      se_s[idx] = data[base + 2 * (size_t)TDIM * FDIM];
      if (t + 1 < TDIM) __builtin_prefetch(&data[base + FDIM], 0, 1);
    }
    __syncthreads();

    // ---- P1: gamma_h = exp(-relu(d @ W_dh^T + b_dh))  (M=16,N=128,K=36) ----
    {
      int nt = wv;                                   // 8 waves -> 8 N-tiles of H
      v8f acc = {0,0,0,0,0,0,0,0};
      for (int s = 0; s < FDIM / 4; ++s) {
        int k0 = s * 4 + half * 2;
        v2f a = { d_s[l16 * FDIM + k0], d_s[l16 * FDIM + k0 + 1] };
        const float* Bp = wsWdhT + nt * 16 + l16;
        v2f b = { Bp[k0 * HDIM], Bp[(k0 + 1) * HDIM] };
        acc = wmma4(a, b, acc);
      }
      for (int v = 0; v < 8; ++v) {
        int bb = v + half * 8;
        int hc = nt * 16 + l16;
        float val = acc[v] + sb_dh[hc];
        gh[bb * HDIM + hc] = __expf(-fmaxf(val, 0.0f));
      }
    }
    // gamma_x (elementwise, diag(W_dx))
    for (int idx = tid; idx < BB * FDIM; idx += 256) {
      int f = idx % FDIM;
      gx[idx] = __expf(-fmaxf(d_s[idx] * swdx[f] + sbdx[f], 0.0f));
    }
    __syncthreads();

    // ---- P2: decay h in place ----
    for (int idx = tid; idx < RROWS * HDIM; idx += 256) {
      int row = idx / HDIM, hc = idx % HDIM;
      hR[idx] *= gh[(row / K3) * HDIM + hc];
    }
    __syncthreads();

    // ---- P3: x_h = h @ W_hist^T + b_hist  (M=48,N=36->48,K=128) ----
    for (int t9 = wv; t9 < 9; t9 += 8) {
      int mt = t9 / 3, nt = t9 % 3;
      v8f acc = {0,0,0,0,0,0,0,0};
      int arow = mt * 16 + l16;
      int col = nt * 16 + l16;
      for (int s = 0; s < HDIM / 4; ++s) {
        int k0 = s * 4 + half * 2;
        v2f a = { hR[arow * HDIM + k0], hR[arow * HDIM + k0 + 1] };
        v2f b = { 0.0f, 0.0f };
        if (col < FDIM) { b.x = wsWhistT[k0 * FDIM + col]; b.y = wsWhistT[(k0 + 1) * FDIM + col]; }
        acc = wmma4(a, b, acc);
      }
      for (int v = 0; v < 8; ++v) {
        int row = mt * 16 + v + half * 8;
        if (col < FDIM) xh_s[row * FDIM + col] = acc[v] + sb_hist[col];
      }
    }
    __syncthreads();

    // ---- P4: loss1, x_c, denom partial ----
    for (int idx = tid; idx < RROWS * FDIM; idx += 256) {
      int row = idx / FDIM, f = idx % FDIM;
      int bb = row / K3, k = row % K3;
      float m = m_s[bb * FDIM + f];
      float rv = rt_s[idx], xh = xh_s[idx];
      float wkv = (k == 0) ? wk0 : ((k == 1) ? wk1 : wk2);
      pl1 += wkv * fabsf(rv - xh) * m;
      xc_s[idx] = m * rv + (1.0f - m) * xh;
    }
    for (int idx = tid; idx < BB * FDIM; idx += 256) pden += wksum * m_s[idx];
    __syncthreads();

    // ---- P5: z_h = x_c @ Wf_masked^T + b_feat  (M=48,N=36->48,K=36) ----
    for (int t9 = wv; t9 < 9; t9 += 8) {
      int mt = t9 / 3, nt = t9 % 3;
      v8f acc = {0,0,0,0,0,0,0,0};
      int arow = mt * 16 + l16, col = nt * 16 + l16;
      for (int s = 0; s < FDIM / 4; ++s) {
        int k0 = s * 4 + half * 2;
        v2f a = { xc_s[arow * FDIM + k0], xc_s[arow * FDIM + k0 + 1] };
        v2f b = { 0.0f, 0.0f };
        if (col < FDIM) { b.x = wsWfT[k0 * FDIM + col]; b.y = wsWfT[(k0 + 1) * FDIM + col]; }
        acc = wmma4(a, b, acc);
      }
      for (int v = 0; v < 8; ++v) {
        int row = mt * 16 + v + half * 8;
        if (col < FDIM) zh_s[row * FDIM + col] = acc[v] + sb_feat[col];
      }
    }
    __syncthreads();

    // ---- P6: alpha = [gamma_x, m] @ W_wc^T + b_wc  (M=48,N=36->48,K=72) ----
    for (int t9 = wv; t9 < 9; t9 += 8) {
      int mt = t9 / 3, nt = t9 % 3;
      v8f acc = {0,0,0,0,0,0,0,0};
      int arow = mt * 16 + l16, bb = arow / K3, col = nt * 16 + l16;
      for (int s = 0; s < (2 * FDIM) / 4; ++s) {
        int k0 = s * 4 + half * 2;
        float ax = (k0 < FDIM) ? gx[bb * FDIM + k0] : m_s[bb * FDIM + k0 - FDIM];
        float ay = (k0 + 1 < FDIM) ? gx[bb * FDIM + k0 + 1] : m_s[bb * FDIM + k0 + 1 - FDIM];
        v2f a = { ax, ay };
        v2f b = { 0.0f, 0.0f };
        if (col < FDIM) { b.x = wsWwcT[k0 * FDIM + col]; b.y = wsWwcT[(k0 + 1) * FDIM + col]; }
        acc = wmma4(a, b, acc);
      }
      for (int v = 0; v < 8; ++v) {
        int row = mt * 16 + v + half * 8;
        if (col < FDIM) al_s[row * FDIM + col] = acc[v] + sb_wc[col];
      }
    }
    __syncthreads();

    // ---- P7a: loss2/loss3, c_h, c_c ----
    for (int idx = tid; idx < RROWS * FDIM; idx += 256) {
      int row = idx / FDIM, f = idx % FDIM;
      int bb = row / K3, k = row % K3;
      float m = m_s[bb * FDIM + f], rv = rt_s[idx];
      float wkv = (k == 0) ? wk0 : ((k == 1) ? wk1 : wk2);
      float zh = zh_s[idx], xh = xh_s[idx], al = al_s[idx];
      pl2 += wkv * fabsf(rv - zh) * m;
      float chv = al * zh + (1.0f - al) * xh;
      pl3 += wkv * fabsf(rv - chv) * m;
      cc_s[idx] = m * rv + (1.0f - m) * chv;
    }
    __syncthreads();

    // ---- P7b: imputations out ----
    for (int idx = tid; idx < BB * FDIM; idx += 256) {
      int bb = idx / FDIM, f = idx % FDIM;
      int b = blk * BB + bb;
      float acc =
          wk0 * (cc_s[(bb * K3 + 0) * FDIM + f] + se_s[(bb * K3 + 0) * FDIM + f] + tr_s[(bb * K3 + 0) * FDIM + f]) +
          wk1 * (cc_s[(bb * K3 + 1) * FDIM + f] + se_s[(bb * K3 + 1) * FDIM + f] + tr_s[(bb * K3 + 1) * FDIM + f]) +
          wk2 * (cc_s[(bb * K3 + 2) * FDIM + f] + se_s[(bb * K3 + 2) * FDIM + f] + tr_s[(bb * K3 + 2) * FDIM + f]);
      out_imp[((size_t)b * TDIM + t) * FDIM + f] = acc;
    }

    // ---- P8: gates GEMM (M=48,N=512,K=200) fused with LSTM cell ----
    // Each wave owns a (16-row x 16-hcol) quadruple: gate tiles i/f/g/o in regs.
    for (int q = wv; q < 24; q += 8) {
      int mt = q >> 3, ht = q & 7;
      v8f a0 = {0,0,0,0,0,0,0,0}, a1 = {0,0,0,0,0,0,0,0};
      v8f a2 = {0,0,0,0,0,0,0,0}, a3 = {0,0,0,0,0,0,0,0};
      int arow = mt * 16 + l16;
      int bb = arow / K3;
      const float* Bp = wsWT + ht * 16 + l16;
      for (int s = 0; s < KIN / 4; ++s) {
        int k0 = s * 4 + half * 2;
        int k1 = k0 + 1;
        float ax = (k0 < FDIM) ? cc_s[arow * FDIM + k0]
                 : (k0 < 2 * FDIM) ? m_s[bb * FDIM + k0 - FDIM]
                                   : hR[arow * HDIM + k0 - 2 * FDIM];
        float ay = (k1 < FDIM) ? cc_s[arow * FDIM + k1]
                 : (k1 < 2 * FDIM) ? m_s[bb * FDIM + k1 - FDIM]
                                   : hR[arow * HDIM + k1 - 2 * FDIM];
        v2f a = { ax, ay };
        v2f b0 = { Bp[k0 * H4 +   0], Bp[k1 * H4 +   0] };
        v2f b1 = { Bp[k0 * H4 + 128], Bp[k1 * H4 + 128] };
        v2f b2 = { Bp[k0 * H4 + 256], Bp[k1 * H4 + 256] };
        v2f b3 = { Bp[k0 * H4 + 384], Bp[k1 * H4 + 384] };
        a0 = wmma4(a, b0, a0);
        a1 = wmma4(a, b1, a1);
        a2 = wmma4(a, b2, a2);
        a3 = wmma4(a, b3, a3);
      }
      int hcol = ht * 16 + l16;
      float bI = sb_ih[hcol]       + sb_hh[hcol];
      float bF = sb_ih[128 + hcol] + sb_hh[128 + hcol];
      float bG = sb_ih[256 + hcol] + sb_hh[256 + hcol];
      float bO = sb_ih[384 + hcol] + sb_hh[384 + hcol];
      for (int v = 0; v < 8; ++v) {
        int row = mt * 16 + v + half * 8;
        float iv = sigf(a0[v] + bI);
        float fv = sigf(a1[v] + bF);
        float gv = tanhf(a2[v] + bG);
        float ov = sigf(a3[v] + bO);
        float cn = fv * c_s[row * HDIM + hcol] + iv * gv;
        c_s[row * HDIM + hcol] = cn;
        hW[row * HDIM + hcol] = ov * tanhf(cn);
      }
    }
    __syncthreads();

    // ---- P9: per-step loss partial reduction -> ws ----
    {
      float vals[4] = { pl1, pl2, pl3, pden };
      for (int i = 0; i < 4; ++i) {
        float v = vals[i];
        for (int off = 16; off > 0; off >>= 1) v += __shfl_down(v, off, 32);
        if (lane == 0) red_s[wv * 4 + i] = v;
      }
      __syncthreads();
      if (tid == 0) {
        float s0 = 0, s1 = 0, s2 = 0, s3 = 0;
        for (int w2 = 0; w2 < 8; ++w2) {
          s0 += red_s[w2 * 4 + 0]; s1 += red_s[w2 * 4 + 1];
          s2 += red_s[w2 * 4 + 2]; s3 += red_s[w2 * 4 + 3];
        }
        float* lp = lossPart + ((size_t)t * NBLK + blk) * 4;
        lp[0] = s0; lp[1] = s1; lp[2] = s2; lp[3] = s3;
      }
      __syncthreads();
    }
  }
}

// ---------------------------------------------------------------------------
// Kernel 2: finalize scalar loss
// ---------------------------------------------------------------------------
__global__ void rits_loss_kernel(const float* __restrict__ lossPart,
                                 const float* __restrict__ W_comb,
                                 float* __restrict__ out) {
  __shared__ float red[256];
  int tid = threadIdx.x;
  float c0 = W_comb[0], c1 = W_comb[1], c2 = W_comb[2];
  float mx = fmaxf(c0, fmaxf(c1, c2));
  float e0 = __expf(c0 - mx), e1 = __expf(c1 - mx), e2 = __expf(c2 - mx);
  float es = e0 + e1 + e2;
  float w0 = e0 / es, w1 = e1 / es, w2 = e2 / es;
  float reg = 0.1f * (w0 * (1.0f / 24.0f) + w1 * (1.0f / 168.0f) + w2 * (1.0f / 720.0f));
  float acc = 0.0f;
  for (int t = tid; t < TDIM; t += 256) {
    float l1 = 0, l2 = 0, l3 = 0, den = 1e-5f;
    for (int b2 = 0; b2 < NBLK; ++b2) {
      const float* p = lossPart + ((size_t)t * NBLK + b2) * 4;
      l1 += p[0]; l2 += p[1]; l3 += p[2]; den += p[3];
    }
    acc += (l1 + l2 + l3) / den + reg;
  }
  red[tid] = acc;
  __syncthreads();
  for (int s = 128; s > 0; s >>= 1) {
    if (tid < s) red[tid] += red[tid + s];
    __syncthreads();
  }
  if (tid == 0) out[0] = red[0];
}

extern "C" void kernel_launch(void* const* d_in, const int* in_sizes, int n_in,
                              void* d_out, int out_size, void* d_ws, size_t ws_size,
                              hipStream_t stream) {
  (void)in_sizes; (void)n_in; (void)out_size; (void)ws_size;
  const float* data   = (const float*)d_in[0];
  const float* W_dh   = (const float*)d_in[1];
  const float* b_dh   = (const float*)d_in[2];
  const float* W_dx   = (const float*)d_in[3];
  const float* b_dx   = (const float*)d_in[4];
  const float* W_hist = (const float*)d_in[5];
  const float* b_hist = (const float*)d_in[6];
  const float* W_feat = (const float*)d_in[7];
  const float* b_feat = (const float*)d_in[8];
  const float* W_wc   = (const float*)d_in[9];
  const float* b_wc   = (const float*)d_in[10];
  const float* W_ih   = (const float*)d_in[11];
  const float* W_hh   = (const float*)d_in[12];
  const float* b_ih   = (const float*)d_in[13];
  const float* b_hh   = (const float*)d_in[14];
  const float* W_comb = (const float*)d_in[15];
  float* ws  = (float*)d_ws;
  float* out = (float*)d_out;

  rits_prep_kernel<<<dim3((KIN * H4 + 255) / 256), dim3(256), 0, stream>>>(
      W_dh, W_hist, W_feat, W_wc, W_ih, W_hh, ws);
  rits_scan_kernel<<<dim3(NBLK), dim3(256), 0, stream>>>(
      data, b_dh, W_dx, b_dx, b_hist, b_feat, b_wc, b_ih, b_hh, W_comb,
      ws, ws + WS_LOSS, out + 1);
  rits_loss_kernel<<<dim3(1), dim3(256), 0, stream>>>(ws + WS_LOSS, W_comb, out);
}